// SGC_66915590472493
// MI455X (gfx1250) — compile-verified
//
#include <hip/hip_runtime.h>
#include <math.h>

// ---------------------------------------------------------------------------
// SGC forward for MI455X (gfx1250, wave32).
//   deg/dinv -> coef -> 2x (selfloop-init + edge scatter-add) -> WMMA GEMM+ELU
// The aggregation is L2-resident data movement (x, h1, h2 are 12.8MB each vs
// 192MB L2); the dense projection uses V_WMMA_F32_16X16X4_F32 at full fp32.
// ---------------------------------------------------------------------------

typedef __attribute__((ext_vector_type(2))) float v2f;
typedef __attribute__((ext_vector_type(8))) float v8f;

static __host__ __device__ inline int cdiv(int a, int b) { return (a + b - 1) / b; }

// deg[i] = 1.0f  (self-loop contributes weight 1 to every node's degree)
__global__ void sgc_fill_one(float* __restrict__ deg, int n) {
    int i = blockIdx.x * blockDim.x + threadIdx.x;
    if (i < n) deg[i] = 1.0f;
}

// deg[col[e]] += w[e]
__global__ void sgc_deg_accum(float* __restrict__ deg,
                              const long long* __restrict__ col,
                              const float* __restrict__ w, int E) {
    int e = blockIdx.x * blockDim.x + threadIdx.x;
    if (e < E) atomicAdd(&deg[(int)col[e]], w[e]);
}

// in-place: deg -> dinv = deg > 0 ? rsqrt(deg) : 0
__global__ void sgc_rsqrt(float* __restrict__ deg, int n) {
    int i = blockIdx.x * blockDim.x + threadIdx.x;
    if (i < n) {
        float d = deg[i];
        deg[i] = (d > 0.0f) ? rsqrtf(d) : 0.0f;
    }
}

// coef[e] = dinv[row[e]] * w[e] * dinv[col[e]]   (reused by both hops)
__global__ void sgc_coef(float* __restrict__ coef,
                         const float* __restrict__ dinv,
                         const long long* __restrict__ row,
                         const long long* __restrict__ col,
                         const float* __restrict__ w, int E) {
    int e = blockIdx.x * blockDim.x + threadIdx.x;
    if (e < E) coef[e] = dinv[(int)row[e]] * w[e] * dinv[(int)col[e]];
}

// dst[i] = dinv[node]^2 * src[i]  -- self-loop term doubles as the zero-init
// of the hop destination (saves a separate memset + add pass).
__global__ void sgc_selfloop_init(const float* __restrict__ src,
                                  float* __restrict__ dst,
                                  const float* __restrict__ dinv, int total) {
    int i = blockIdx.x * blockDim.x + threadIdx.x;
    if (i < total) {
        float d = dinv[i >> 6];  // F == 64
        dst[i] = d * d * src[i];
    }
}

// One 16-lane group per edge; each lane moves a float4 (4 features).
// Gather of the 256B source row is one coalesced burst; the scatter is 4
// global_atomic_add_f32 per lane resolving in L2 (dst fits in 192MB L2).
__global__ void sgc_hop(const float* __restrict__ src, float* __restrict__ dst,
                        const long long* __restrict__ row,
                        const long long* __restrict__ col,
                        const float* __restrict__ coef, int E) {
    int g = blockIdx.x * blockDim.x + threadIdx.x;
    int e = g >> 4;
    int l = g & 15;
    if (e >= E) return;
    int r = (int)row[e];
    int c = (int)col[e];
    float cf = coef[e];
    const float4 v = *(const float4*)(src + (size_t)r * 64 + l * 4);
    float* d = dst + (size_t)c * 64 + l * 4;
    atomicAdd(d + 0, cf * v.x);
    atomicAdd(d + 1, cf * v.y);
    atomicAdd(d + 2, cf * v.z);
    atomicAdd(d + 3, cf * v.w);
}

// out[m,n] = ELU( sum_k h[m,k] * W[n,k] + b[n] )
// One wave computes a 16x64 tile via 64x V_WMMA_F32_16X16X4_F32 (full fp32).
// A 16x4 layout : lanes 0-15 hold M, VGPR0/1 = K(kp)/K(kp+1), kp = (lane>>4)*2.
// B 4x16 layout : lane&15 = N, same lane-half K split (mirrors A).
// C/D 16x16     : lane&15 = N, VGPR r = M = r + 8*(lane>>4).
__global__ void sgc_gemm_bias_elu(const float* __restrict__ h,
                                  const float* __restrict__ Wg,
                                  const float* __restrict__ bg,
                                  float* __restrict__ out,
                                  int ntiles, int N) {
    __shared__ float Ws[64 * 64];
    __shared__ float bs[64];
    for (int i = threadIdx.x; i < 64 * 64; i += blockDim.x) Ws[i] = Wg[i];
    if (threadIdx.x < 64) bs[threadIdx.x] = bg[threadIdx.x];
    __syncthreads();

    const int wave = threadIdx.x >> 5;
    const int lane = threadIdx.x & 31;
    const int tile = blockIdx.x * 8 + wave;
    if (tile >= ntiles) return;  // uniform per wave -> EXEC stays all-ones

    const int m0  = tile * 16;
    const int l15 = lane & 15;          // A: M within tile; B/C: N within tile
    const int kp  = (lane >> 4) * 2;    // K-pair selector for A and B
    int mA = m0 + l15; if (mA >= N) mA = N - 1;   // clamp (N%16==0 normally)
    const float* hrow = h + (size_t)mA * 64;

    v8f acc[4];
#pragma unroll
    for (int nt = 0; nt < 4; ++nt) {
        v8f c = {};
        const float* wrow = Ws + (nt * 16 + l15) * 64;  // W[n,:] row (= B column n)
#pragma unroll
        for (int kk = 0; kk < 64; kk += 4) {
            v2f a, b;
            a.x = hrow[kk + kp];
            a.y = hrow[kk + kp + 1];
            b.x = wrow[kk + kp];
            b.y = wrow[kk + kp + 1];
            // 8 args: (neg_a, A, neg_b, B, c_mod, C, reuse_a, reuse_b)
            c = __builtin_amdgcn_wmma_f32_16x16x4_f32(
                false, a, false, b, (short)0, c, false, false);
        }
        acc[nt] = c;
    }

    const int mhi = (lane >> 4) * 8;
#pragma unroll
    for (int nt = 0; nt < 4; ++nt) {
        const int n0 = nt * 16;
        const float bval = bs[n0 + l15];
#pragma unroll
        for (int r = 0; r < 8; ++r) {
            const int m = m0 + mhi + r;
            if (m < N) {
                float v = acc[nt][r] + bval;
                v = (v > 0.0f) ? v : expm1f(v);  // ELU, alpha=1
                out[(size_t)m * 64 + n0 + l15] = v;
            }
        }
    }
}

extern "C" void kernel_launch(void* const* d_in, const int* in_sizes, int n_in,
                              void* d_out, int out_size, void* d_ws, size_t ws_size,
                              hipStream_t stream) {
    const float* x          = (const float*)d_in[0];         // [N,64] f32
    const long long* ei     = (const long long*)d_in[1];     // [2,E] int64
    const float* w          = (const float*)d_in[2];         // [E]   f32
    const float* W          = (const float*)d_in[3];         // [64,64] f32
    const float* b          = (const float*)d_in[4];         // [64]  f32
    float* out              = (float*)d_out;                 // [N,64] f32

    const int F = 64;
    const int N = in_sizes[0] / F;
    const int E = in_sizes[2];
    const long long* row = ei;       // source
    const long long* col = ei + E;   // target

    // workspace: deg/dinv[N] | coef[E] | h1[N*F] | h2[N*F]  (~31 MB)
    float* ws   = (float*)d_ws;
    float* dinv = ws;
    float* coef = ws + N;
    float* h1   = coef + E;
    float* h2   = h1 + (size_t)N * F;

    const int B = 256;

    // --- gcn_norm ---
    sgc_fill_one<<<cdiv(N, B), B, 0, stream>>>(dinv, N);              // deg = 1 (self loop)
    sgc_deg_accum<<<cdiv(E, B), B, 0, stream>>>(dinv, col, w, E);     // deg += w
    sgc_rsqrt<<<cdiv(N, B), B, 0, stream>>>(dinv, N);                 // deg -> dinv
    sgc_coef<<<cdiv(E, B), B, 0, stream>>>(coef, dinv, row, col, w, E);

    // --- hop 1: h1 = dinv^2*x (self loop) + scatter(coef * x[row] -> col) ---
    sgc_selfloop_init<<<cdiv(N * F, B), B, 0, stream>>>(x, h1, dinv, N * F);
    sgc_hop<<<cdiv(E * 16, B), B, 0, stream>>>(x, h1, row, col, coef, E);

    // --- hop 2: h2 = dinv^2*h1 + scatter(coef * h1[row] -> col) ---
    sgc_selfloop_init<<<cdiv(N * F, B), B, 0, stream>>>(h1, h2, dinv, N * F);
    sgc_hop<<<cdiv(E * 16, B), B, 0, stream>>>(h1, h2, row, col, coef, E);

    // --- projection + bias + ELU via fp32 WMMA ---
    const int ntiles = cdiv(N, 16);   // 3125 for N=50000 (exact)
    sgc_gemm_bias_elu<<<cdiv(ntiles, 8), 256, 0, stream>>>(h2, W, b, out, ntiles, N);
}